// KimiCognitiveBraid_19576460935948
// MI455X (gfx1250) — compile-verified
//
#include <hip/hip_runtime.h>
#include <math.h>

// N=11 streams, D=1024; per-(b,l) position independent. Memory-bound:
// ~740MB floor -> ~32us @ 23.3 TB/s. All streaming in b128 (NT), dot
// matrices via V_WMMA_F32_16X16X4_F32 (exact f32).

#define NS 11
#define DD 1024
#define TPB 128
#define XPITCH 1028   // padded LDS row stride (words): %64==4 (conflict-free), %4==0 (b128 ok)

typedef __attribute__((ext_vector_type(2))) float v2f;
typedef __attribute__((ext_vector_type(4))) float f4;
typedef __attribute__((ext_vector_type(8))) float v8f;

__device__ inline float wave_reduce_add(float v) {
#pragma unroll
  for (int off = 16; off > 0; off >>= 1)
    v += __shfl_down(v, off, 32);
  return v;
}

// Precompute qg = q*gamma (per row), S = sum(qg), qb = q·beta for attn & mlp.
__global__ void braid_setup(const float* __restrict__ aq, const float* __restrict__ ag,
                            const float* __restrict__ ab, const float* __restrict__ mq,
                            const float* __restrict__ mg, const float* __restrict__ mb,
                            float* __restrict__ qgA, float* __restrict__ qgM,
                            float* __restrict__ scal) {
  const int i = blockIdx.x;
  const int t = threadIdx.x;
  float sA = 0.f, bA = 0.f, sM = 0.f, bM = 0.f;
  for (int d = t; d < DD; d += 256) {
    const float qa = aq[i * DD + d], qm = mq[i * DD + d];
    const float va = qa * ag[i * DD + d];
    const float vm = qm * mg[i * DD + d];
    qgA[i * DD + d] = va;
    qgM[i * DD + d] = vm;
    sA += va; sM += vm;
    bA += qa * ab[i * DD + d];
    bM += qm * mb[i * DD + d];
  }
  __shared__ float red[8][4];
  const int lane = t & 31, wv = t >> 5;
  sA = wave_reduce_add(sA);
  bA = wave_reduce_add(bA);
  sM = wave_reduce_add(sM);
  bM = wave_reduce_add(bM);
  if (lane == 0) { red[wv][0] = sA; red[wv][1] = bA; red[wv][2] = sM; red[wv][3] = bM; }
  __syncthreads();
  if (t == 0) {
    float a = 0.f, b = 0.f, c = 0.f, e = 0.f;
    for (int w = 0; w < 8; ++w) { a += red[w][0]; b += red[w][1]; c += red[w][2]; e += red[w][3]; }
    scal[i] = a; scal[16 + i] = b; scal[32 + i] = c; scal[48 + i] = e;
  }
}

__global__ __launch_bounds__(TPB) void braid_main(
    const float* __restrict__ streams, const float* __restrict__ qgA,
    const float* __restrict__ qgM, const float* __restrict__ scal,
    float* __restrict__ out, int P) {
  __shared__ __align__(16) float xs[NS * XPITCH];  // 11 stream vectors
  __shared__ float wred[4][2][256];                // per-wave WMMA partials
  __shared__ float G[256];                         // attn dots  G[i*16+k]
  __shared__ float Mt[256];                        // mlp  dots  Mt[i*16+k]
  __shared__ float wsum1[16][4];
  __shared__ float wsum2[16][4];
  __shared__ float mu_s[16];
  __shared__ float rstd_s[16];
  __shared__ float hred[4][4];
  __shared__ float hstat[4];

  const int p = blockIdx.x;            // position index in [0, B*L)
  const int t = threadIdx.x;
  const int lane = t & 31;
  const int wave = t >> 5;
  const size_t kstride = (size_t)P * DD;
  const float* xg = streams + (size_t)p * DD;

  // ---- Stream this position's 11 D-vectors into LDS as b128 (read-once: NT)
#pragma unroll
  for (int k = 0; k < NS; ++k) {
#pragma unroll
    for (int j = 0; j < 2; ++j) {
      const int base = 4 * t + 512 * j;
      const f4 v = __builtin_nontemporal_load(
          (const f4*)(xg + (size_t)k * kstride + base));
      *(f4*)(&xs[k * XPITCH + base]) = v;
    }
  }
  __syncthreads();

  // ---- Per-block LayerNorm stats (biased variance, matches jnp.var)
  for (int k = 0; k < NS; ++k) {
    f4 s4 = {0.f, 0.f, 0.f, 0.f}, q4 = {0.f, 0.f, 0.f, 0.f};
#pragma unroll
    for (int j = 0; j < 2; ++j) {
      const f4 v = *(const f4*)(&xs[k * XPITCH + 4 * t + 512 * j]);
      s4 += v;
      q4 += v * v;
    }
    float s = (s4[0] + s4[1]) + (s4[2] + s4[3]);
    float s2 = (q4[0] + q4[1]) + (q4[2] + q4[3]);
    s = wave_reduce_add(s);
    s2 = wave_reduce_add(s2);
    if (lane == 0) { wsum1[k][wave] = s; wsum2[k][wave] = s2; }
  }
  __syncthreads();
  if (t < NS) {
    const float s = wsum1[t][0] + wsum1[t][1] + wsum1[t][2] + wsum1[t][3];
    const float s2 = wsum2[t][0] + wsum2[t][1] + wsum2[t][2] + wsum2[t][3];
    const float mu = s * (1.f / DD);
    const float var = s2 * (1.f / DD) - mu * mu;
    mu_s[t] = mu;
    rstd_s[t] = 1.f / sqrtf(var + 1e-5f);
  }
  __syncthreads();

  // ---- Dot matrices via V_WMMA_F32_16X16X4_F32, K = D split across 4 waves.
  // A (16x4 f32): lane row = lane&15, VGPR0/1 = K = 2*(lane>>4) + {0,1}
  // B (4x16 f32): lane col = lane&15, VGPR0/1 = K = 2*(lane>>4) + {0,1}
  // Rows/cols >= NS are clamped to row 0 (branch-free); their outputs are
  // never read, so no zero-padding is needed.
  {
    const int row = lane & 15;
    const int hh = lane >> 4;
    const int rowc = (row < NS) ? row : 0;
    const float* aptr = qgA + rowc * DD + 2 * hh;
    const float* mptr = qgM + rowc * DD + 2 * hh;
    const float* xptr = xs + rowc * XPITCH + 2 * hh;
    v8f accA = {0.f, 0.f, 0.f, 0.f, 0.f, 0.f, 0.f, 0.f};
    v8f accM = {0.f, 0.f, 0.f, 0.f, 0.f, 0.f, 0.f, 0.f};
    const int d0 = wave * (DD / 4);
#pragma unroll 4
    for (int dd = d0; dd < d0 + DD / 4; dd += 4) {
      v2f a, m, b;
      a[0] = aptr[dd]; a[1] = aptr[dd + 1];
      m[0] = mptr[dd]; m[1] = mptr[dd + 1];
      b[0] = xptr[dd]; b[1] = xptr[dd + 1];
      accA = __builtin_amdgcn_wmma_f32_16x16x4_f32(false, a, false, b, (short)0, accA,
                                                   false, false);
      accM = __builtin_amdgcn_wmma_f32_16x16x4_f32(false, m, false, b, (short)0, accM,
                                                   false, false);
    }
    // C/D layout: VGPR r -> m = r + 8*(lane>>4), n = lane&15
#pragma unroll
    for (int r = 0; r < 8; ++r) {
      const int e = (r + 8 * hh) * 16 + row;
      wred[wave][0][e] = accA[r];
      wred[wave][1][e] = accM[r];
    }
  }
  __syncthreads();
  for (int e = t; e < 256; e += TPB) {
    G[e]  = wred[0][0][e] + wred[1][0][e] + wred[2][0][e] + wred[3][0][e];
    Mt[e] = wred[0][1][e] + wred[1][1][e] + wred[2][1][e] + wred[3][1][e];
  }
  __syncthreads();

  // ---- Sequential over the 11 steps
  for (int i = 0; i < NS; ++i) {
    const float SAi = scal[i];
    const float QBAi = scal[16 + i];
    const float SMi = scal[32 + i];
    const float QBMi = scal[48 + i];

    // first attention: softmax over k=0..i (redundant per thread, cheap)
    float w[NS];
    float mx = -3.0e38f;
#pragma unroll
    for (int k = 0; k < NS; ++k) {
      if (k <= i) {
        const float lg = rstd_s[k] * (G[i * 16 + k] - mu_s[k] * SAi) + QBAi;
        w[k] = lg;
        mx = fmaxf(mx, lg);
      }
    }
    float se = 0.f;
#pragma unroll
    for (int k = 0; k < NS; ++k) {
      if (k <= i) { const float e = expf(w[k] - mx); w[k] = e; se += e; }
    }
    const float inv = 1.f / se;

    // h1 = weighted sum; also accumulate its LN stats and qg_mlp[i]·h1
    f4 h1[2];
    float sh = 0.f, sh2 = 0.f, sdot = 0.f;
#pragma unroll
    for (int j = 0; j < 2; ++j) {
      const int base = 4 * t + 512 * j;
      f4 acc = {0.f, 0.f, 0.f, 0.f};
#pragma unroll
      for (int k = 0; k < NS; ++k)
        if (k <= i) acc += w[k] * *(const f4*)(&xs[k * XPITCH + base]);
      acc *= inv;
      h1[j] = acc;
      const f4 qm = *(const f4*)(qgM + i * DD + base);
      sh += (acc[0] + acc[1]) + (acc[2] + acc[3]);
      const f4 a2 = acc * acc;
      sh2 += (a2[0] + a2[1]) + (a2[2] + a2[3]);
      const f4 dt = qm * acc;
      sdot += (dt[0] + dt[1]) + (dt[2] + dt[3]);
    }
    sh = wave_reduce_add(sh);
    sh2 = wave_reduce_add(sh2);
    sdot = wave_reduce_add(sdot);
    if (lane == 0) { hred[wave][0] = sh; hred[wave][1] = sh2; hred[wave][2] = sdot; }
    __syncthreads();
    if (t == 0) {
      const float a = hred[0][0] + hred[1][0] + hred[2][0] + hred[3][0];
      const float b = hred[0][1] + hred[1][1] + hred[2][1] + hred[3][1];
      const float c = hred[0][2] + hred[1][2] + hred[2][2] + hred[3][2];
      const float mu = a * (1.f / DD);
      const float var = b * (1.f / DD) - mu * mu;
      hstat[0] = mu;
      hstat[1] = 1.f / sqrtf(var + 1e-5f);
      hstat[2] = c;
    }
    __syncthreads();
    const float mu_h = hstat[0], rstd_h = hstat[1], dot_h = hstat[2];

    // second attention: k<i use original blocks, k==i uses h1
    float w2[NS];
    float wh = rstd_h * (dot_h - mu_h * SMi) + QBMi;
    float mx2 = wh;
#pragma unroll
    for (int k = 0; k < NS; ++k) {
      if (k < i) {
        const float lg = rstd_s[k] * (Mt[i * 16 + k] - mu_s[k] * SMi) + QBMi;
        w2[k] = lg;
        mx2 = fmaxf(mx2, lg);
      }
    }
    const float wh_e = expf(wh - mx2);
    float se2 = wh_e;
#pragma unroll
    for (int k = 0; k < NS; ++k) {
      if (k < i) { const float e = expf(w2[k] - mx2); w2[k] = e; se2 += e; }
    }
    const float inv2 = 1.f / se2;

#pragma unroll
    for (int j = 0; j < 2; ++j) {
      const int base = 4 * t + 512 * j;
      f4 acc = wh_e * h1[j];
#pragma unroll
      for (int k = 0; k < NS; ++k)
        if (k < i) acc += w2[k] * *(const f4*)(&xs[k * XPITCH + base]);
      acc *= inv2;
      __builtin_nontemporal_store(
          acc, (f4*)(out + (size_t)i * kstride + (size_t)p * DD + base));
    }
  }
}

extern "C" void kernel_launch(void* const* d_in, const int* in_sizes, int n_in,
                              void* d_out, int out_size, void* d_ws, size_t ws_size,
                              hipStream_t stream) {
  (void)n_in; (void)out_size; (void)ws_size;
  const float* streams = (const float*)d_in[0];
  const float* attn_q  = (const float*)d_in[1];
  const float* attn_g  = (const float*)d_in[2];
  const float* attn_b  = (const float*)d_in[3];
  const float* mlp_q   = (const float*)d_in[4];
  const float* mlp_g   = (const float*)d_in[5];
  const float* mlp_b   = (const float*)d_in[6];

  float* qgA  = (float*)d_ws;           // [NS, D]
  float* qgM  = qgA + NS * DD;          // [NS, D]
  float* scal = qgM + NS * DD;          // [64]: S_attn | qb_attn | S_mlp | qb_mlp
  // ws usage: (2*NS*DD + 64) * 4 bytes ~= 90.4 KB

  const int P = in_sizes[0] / (NS * DD);  // B*L positions

  braid_setup<<<NS, 256, 0, stream>>>(attn_q, attn_g, attn_b, mlp_q, mlp_g, mlp_b,
                                      qgA, qgM, scal);
  braid_main<<<P, TPB, 0, stream>>>(streams, qgA, qgM, scal, (float*)d_out, P);
}